// CTC_85177791414753
// MI455X (gfx1250) — compile-verified
//
#include <hip/hip_runtime.h>
#include <cstdint>

#define Bsz 32
#define Tsz 2048
#define Vsz 1024
#define Lsz 64
#define GST 68            // padded row stride for gathered log-probs (65 used)
#define NEGV (-1.0e30f)

#if defined(__has_builtin)
#  if __has_builtin(__builtin_amdgcn_global_load_async_to_lds_b128) && \
      __has_builtin(__builtin_amdgcn_s_wait_asynccnt)
#    define ASYNC_OK 1
#  endif
#endif
#ifndef ASYNC_OK
#  define ASYNC_OK 0
#endif

#define AS1 __attribute__((address_space(1)))
#define AS3 __attribute__((address_space(3)))
typedef int v4i __attribute__((ext_vector_type(4)));

// copy 16 bytes global -> LDS (async on gfx1250), default (RT) temporal hint
__device__ __forceinline__ void async_cp16(const float* gsrc, float* ldst) {
#if ASYNC_OK
  __builtin_amdgcn_global_load_async_to_lds_b128((AS1 v4i*)gsrc, (AS3 v4i*)ldst, 0, 0);
#else
  *(float4*)ldst = *(const float4*)gsrc;
#endif
}
// streaming variant: TH_NT (cpol bits[2:0] = 1) — evict-early, keeps L2 for g
__device__ __forceinline__ void async_cp16_nt(const float* gsrc, float* ldst) {
#if ASYNC_OK
  __builtin_amdgcn_global_load_async_to_lds_b128((AS1 v4i*)gsrc, (AS3 v4i*)ldst, 0, 1);
#else
  *(float4*)ldst = *(const float4*)gsrc;
#endif
}
__device__ __forceinline__ void async_wait() {
#if ASYNC_OK
  __builtin_amdgcn_s_wait_asynccnt(0);
#endif
  asm volatile("" ::: "memory");
}

__device__ __forceinline__ float lae2(float a, float b) {
  float m = fmaxf(a, b);
  return m + __logf(__expf(a - m) + __expf(b - m));
}
__device__ __forceinline__ float lae3(float a, float b, float c) {
  float m = fmaxf(fmaxf(a, b), c);
  return m + __logf(__expf(a - m) + __expf(b - m) + __expf(c - m));
}

// ---------------------------------------------------------------------------
// Pass 1: per (b,t) row of hs_pad: logsumexp over V, gather 65 label values.
// One wave32 per row, 8 rows per 256-thread block. Row staged to LDS via
// non-temporal async global->LDS copy, reduced with wave32 shuffles.
// ---------------------------------------------------------------------------
__global__ __launch_bounds__(256) void ctc_rows(const float* __restrict__ hs,
                                                const int* __restrict__ ys,
                                                float* __restrict__ g) {
  __shared__ __align__(16) float sh[8 * Vsz];
  const int wave = threadIdx.x >> 5;
  const int lane = threadIdx.x & 31;
  const int row  = blockIdx.x * 8 + wave;   // row in [0, B*T)
  const int b    = row >> 11;               // row / T   (T = 2048)
  const float* src = hs + (size_t)row * Vsz;
  float* shr = sh + wave * Vsz;

#pragma unroll
  for (int i = 0; i < 8; ++i) {             // 8 * 32 lanes * 16B = 4KB row
    const int off = i * 128 + lane * 4;
    async_cp16_nt(src + off, shr + off);    // hs is stream-once: NT hint
  }
  async_wait();

  float m = -3.4e38f;
  float4 v[8];
#pragma unroll
  for (int k = 0; k < 8; ++k) {
    v[k] = *(const float4*)(shr + k * 128 + lane * 4);
    m = fmaxf(m, fmaxf(fmaxf(v[k].x, v[k].y), fmaxf(v[k].z, v[k].w)));
  }
#pragma unroll
  for (int d = 16; d >= 1; d >>= 1) m = fmaxf(m, __shfl_xor(m, d, 32));
  float s = 0.f;
#pragma unroll
  for (int k = 0; k < 8; ++k)
    s += __expf(v[k].x - m) + __expf(v[k].y - m) +
         __expf(v[k].z - m) + __expf(v[k].w - m);
#pragma unroll
  for (int d = 16; d >= 1; d >>= 1) s += __shfl_xor(s, d, 32);
  const float lse = m + __logf(s);

  // gather: g[row][0] = blank logprob; g[row][1+j] = logprob of ys[b][j]
  const int* yrow = ys + b * Lsz;
  const int y0 = yrow[lane], y1 = yrow[lane + 32];
  float* grow = g + (size_t)row * GST;
  grow[1 + lane]  = shr[y0] - lse;
  grow[33 + lane] = shr[y1] - lse;
  if (lane == 0) grow[0] = shr[0] - lse;
}

// ---------------------------------------------------------------------------
// Pass 2: forward alpha recursion. One wave32 per batch element.
// Lane l owns states {4l,4l+1,4l+2,4l+3}; lane 31 also owns state 128.
// Neighbor dependency (s-1, s-2) satisfied by one __shfl_up per step.
// Per-step lp rows are double-buffered in LDS (32 time steps per chunk),
// refilled with async global->LDS copies (RT hint: g lives in L2) that
// overlap with compute; synchronized by s_wait_asynccnt.
// ---------------------------------------------------------------------------
__global__ __launch_bounds__(32) void ctc_alpha(const float* __restrict__ g,
                                                const int* __restrict__ hlens,
                                                const int* __restrict__ ys,
                                                const int* __restrict__ ylens,
                                                float* __restrict__ nll_out) {
  __shared__ __align__(16) float sg[2][32 * GST];   // 2 x 8.7 KB
  const int b = blockIdx.x;
  const int lane = threadIdx.x;
  const float* gb = g + (size_t)b * Tsz * GST;

  // allow_skip per odd state: s=4l+1 -> label j0=2l ; s=4l+3 -> label j1=2l+1
  const int* yrow = ys + b * Lsz;
  const int j0 = 2 * lane, j1 = 2 * lane + 1;
  const int yj0 = yrow[j0];
  const int yj1 = yrow[j1];
  const int ym1 = (lane == 0) ? 0 : yrow[j0 - 1];
  const bool skip1 = (lane > 0) && (yj0 != ym1);  // s=1 (lane 0) never skips
  const bool skip3 = (yj1 != yj0);                // j1 >= 1 always

  const int tmax  = hlens[b];       // updates applied for t in [1, tmax-1]
  const int lastT = tmax - 1;
  const int nchunk = lastT / 32 + 1;

  auto copy_chunk = [&](int c, int buf) {
    const float* srcc = gb + (size_t)(c * 32) * GST;
    float* dstc = &sg[buf][0];
#pragma unroll
    for (int i = 0; i < 17; ++i) {            // 32*68 floats = 544 x 16B
      const int off = (i * 32 + lane) * 4;
      async_cp16(srcc + off, dstc + off);
    }
  };

  copy_chunk(0, 0);
  async_wait();

  // t = 0 init: alpha[0..1] = lp0[0..1], rest NEG
  float a0 = NEGV, a1 = NEGV, a2 = NEGV, a3 = NEGV, a4 = NEGV;
  if (lane == 0) { a0 = sg[0][0]; a1 = sg[0][1]; }

  for (int c = 0; c < nchunk; ++c) {
    if (c + 1 < nchunk) copy_chunk(c + 1, (c + 1) & 1);
    const float* base0 = &sg[c & 1][0];
    const int t0 = c * 32;
    const int tb = (c == 0) ? 1 : t0;
    const int te = min(t0 + 31, lastT);
    for (int t = tb; t <= te; ++t) {
      const float* base = base0 + (t - t0) * GST;
      const float lpB = base[0];              // blank (broadcast read)
      const float lp1 = base[2 * lane + 1];
      const float lp2 = base[2 * lane + 2];
      float p1 = __shfl_up(a3, 1, 32);        // state 4l-1 (old)
      if (lane == 0) p1 = NEGV;
      const float n0 = lae2(a0, p1) + lpB;                         // s=4l   (blank)
      const float n1 = lae3(a1, a0, skip1 ? p1 : NEGV) + lp1;      // s=4l+1
      const float n2 = lae2(a2, a1) + lpB;                         // s=4l+2 (blank)
      const float n3 = lae3(a3, a2, skip3 ? a1 : NEGV) + lp2;      // s=4l+3
      const float n4 = (lane == 31) ? (lae2(a4, a3) + lpB) : NEGV; // s=128
      a0 = n0; a1 = n1; a2 = n2; a3 = n3; a4 = n4;
    }
    if (c + 1 < nchunk) async_wait();
  }

  // nll = -logaddexp(alpha[2*olen], alpha[2*olen-1])
  const int olen = ylens[b];
  const int se = 2 * olen;                 // in [64, 128]
  float aend, apre;
  if (se == 128) {
    aend = __shfl(a4, 31, 32);
  } else {
    const int k = se & 3, sl = se >> 2;
    const float x = (k == 0) ? a0 : (k == 1) ? a1 : (k == 2) ? a2 : a3;
    aend = __shfl(x, sl, 32);
  }
  {
    const int sp = se - 1;
    const int k = sp & 3, sl = sp >> 2;
    const float x = (k == 0) ? a0 : (k == 1) ? a1 : (k == 2) ? a2 : a3;
    apre = __shfl(x, sl, 32);
  }
  if (lane == 0) {
    const float nll = -lae2(aend, apre);
    // loss = mean_b(nll/olen)/B  ->  sum_b nll/(olen*B*B)
    nll_out[b] = nll / ((float)olen * (float)(Bsz * Bsz));
  }
}

// ---------------------------------------------------------------------------
// Pass 3: sum the 32 per-batch contributions into the scalar loss.
// ---------------------------------------------------------------------------
__global__ void ctc_final(const float* __restrict__ nll, float* __restrict__ out) {
  float v = nll[threadIdx.x & 31];
#pragma unroll
  for (int d = 16; d >= 1; d >>= 1) v += __shfl_xor(v, d, 32);
  if (threadIdx.x == 0) out[0] = v;
}

extern "C" void kernel_launch(void* const* d_in, const int* in_sizes, int n_in,
                              void* d_out, int out_size, void* d_ws, size_t ws_size,
                              hipStream_t stream) {
  (void)in_sizes; (void)n_in; (void)out_size; (void)ws_size;
  const float* hs    = (const float*)d_in[0];
  const int*   hlens = (const int*)d_in[1];
  const int*   ys    = (const int*)d_in[2];
  const int*   ylens = (const int*)d_in[3];
  float* out = (float*)d_out;
  float* ws  = (float*)d_ws;
  float* nll = ws;            // 32 floats
  float* g   = ws + 1024;     // +4KB; needs ~17.9 MB of workspace

  ctc_rows<<<(Bsz * Tsz) / 8, 256, 0, stream>>>(hs, ys, g);
  ctc_alpha<<<Bsz, 32, 0, stream>>>(g, hlens, ys, ylens, nll);
  ctc_final<<<1, 32, 0, stream>>>(nll, out);
}